// SampleAndAggregate_84963043049898
// MI455X (gfx1250) — compile-verified
//
#include <hip/hip_runtime.h>
#include <math.h>

// ---------------- problem constants (from reference) ----------------
#define NNODE 200000
#define DIM   256          // feature dim == 2*OUT (concat width)
#define BATCH 1024
#define NEGS  64
#define S1N   25
#define S2N   10
#define OUTC  128          // per-layer output dim (half of concat)

#define LDSW  (DIM + 4)    // padded LDS row stride (floats, even -> 8B-aligned pairs)
#define MT    4            // M tiles (of 16 rows) per block
#define MROWS (16 * MT)    // rows staged per block = 64

typedef __attribute__((ext_vector_type(2))) float v2f;
typedef __attribute__((ext_vector_type(8))) float v8f;
typedef int v4i __attribute__((vector_size(4 * sizeof(int))));
typedef __attribute__((address_space(1))) v4i* glob_v4i_p;   // global
typedef __attribute__((address_space(3))) v4i* lds_v4i_p;    // LDS

// ---------------------------------------------------------------
// 16-byte stage into LDS: CDNA5 async global->LDS
// (per-lane gather -> GLOBAL_LOAD_ASYNC_TO_LDS_B128, tracked by ASYNCcnt),
// fallback: plain load+store.
// ---------------------------------------------------------------
__device__ __forceinline__ void stage16(float* lds, const float4* g)
{
#if __has_builtin(__builtin_amdgcn_global_load_async_to_lds_b128)
    __builtin_amdgcn_global_load_async_to_lds_b128(
        (glob_v4i_p)g, (lds_v4i_p)lds, 0, 0);
#else
    *(float4*)lds = *g;
#endif
}

__device__ __forceinline__ void stage_drain()
{
#if __has_builtin(__builtin_amdgcn_global_load_async_to_lds_b128)
#if __has_builtin(__builtin_amdgcn_s_wait_asynccnt)
    __builtin_amdgcn_s_wait_asynccnt(0);
#else
    asm volatile("s_wait_asynccnt 0" ::: "memory");
#endif
#endif
}

// ---------------------------------------------------------------
// LDS-tiled weight transpose: WT[n*DIM + k] = W[k*OUTC + n]
// (K=DIM=256 rows, N=OUTC=128 cols).  Makes the WMMA B fragment a single
// contiguous b64 load per K-step instead of two strided b32 loads.
// ---------------------------------------------------------------
__global__ __launch_bounds__(256)
void transpose_w_kernel(const float* __restrict__ W, float* __restrict__ WT)
{
    __shared__ float tile[16][17];
    const int tx = threadIdx.x & 15;
    const int ty = threadIdx.x >> 4;
    const int k0 = blockIdx.x * 16;            // K tile
    const int n0 = blockIdx.y * 16;            // N tile
    tile[ty][tx] = W[(k0 + ty) * OUTC + n0 + tx];   // coalesced along n
    __syncthreads();
    WT[(n0 + ty) * DIM + k0 + tx] = tile[tx][ty];   // coalesced along k
}

// ---------------------------------------------------------------
// gather + mean over S neighbor rows: out[m,:] = mean_s feat[idx[m*S+s],:]
// idx == nullptr -> contiguous rows (m*S+s). One block per row, 64 lanes,
// float4 (b128) coalesced loads + prefetch of the next row.
// ---------------------------------------------------------------
__global__ __launch_bounds__(64)
void gather_mean_kernel(const float* __restrict__ feat,
                        const int* __restrict__ idx,
                        float* __restrict__ out, int M, int S)
{
    const int m = blockIdx.x;
    const int t = threadIdx.x;                 // float4 column 0..63
    if (m >= M) return;
    float4 acc = make_float4(0.f, 0.f, 0.f, 0.f);
    for (int s = 0; s < S; ++s) {
        long r = idx ? (long)idx[m * S + s] : (long)(m * S + s);
        if (s + 1 < S) {
            long rn = idx ? (long)idx[m * S + s + 1] : (long)(m * S + s + 1);
            __builtin_prefetch(feat + rn * DIM + t * 4, 0, 0);
        }
        float4 v = ((const float4*)(feat + r * DIM))[t];
        acc.x += v.x; acc.y += v.y; acc.z += v.z; acc.w += v.w;
    }
    const float inv = 1.0f / (float)S;
    acc.x *= inv; acc.y *= inv; acc.z *= inv; acc.w *= inv;
    ((float4*)(out + (long)m * DIM))[t] = acc;
}

// ---------------------------------------------------------------
// Fused GraphSAGE concat GEMM:
//   out[m, 0:128]   = act( Aself[m,:]  @ Ws )
//   out[m, 128:256] = act( Aneigh[m,:] @ Wn )
// Weights passed TRANSPOSED: WT[n*DIM + k].
// Block = 512 threads = 16 waves; block covers MROWS=64 rows (4 M-tiles per
// wave) so each B fragment is reused 4x. Waves 0..7: self half; 8..15: neigh
// half. A staged via async global->LDS gather; K marched 4 at a time with
// v_wmma_f32_16x16x4_f32.  Requires M % 64 == 0 (true for all call sites).
// ---------------------------------------------------------------
__global__ __launch_bounds__(512)
void sage_gemm_kernel(const float* __restrict__ featSelf,
                      const int* __restrict__ idxSelf,
                      const float* __restrict__ Aneigh,
                      const float* __restrict__ WsT,
                      const float* __restrict__ WnT,
                      float* __restrict__ out, int M, int doRelu)
{
    extern __shared__ float ldsA[];            // [2][MROWS][LDSW]
    const int m0  = blockIdx.x * MROWS;
    const int tid = threadIdx.x;

    // stage both 64x256 A tiles (b128 per element, async -> LDS)
    for (int i = tid; i < MROWS * (DIM / 4); i += 512) {
        const int row = i / (DIM / 4);
        const int c4  = i % (DIM / 4);
        const int gm  = m0 + row;
        const long rs = idxSelf ? (long)idxSelf[gm] : (long)gm;
        stage16(ldsA + (0 * MROWS + row) * LDSW + c4 * 4,
                (const float4*)(featSelf + rs * DIM) + c4);
        stage16(ldsA + (1 * MROWS + row) * LDSW + c4 * 4,
                (const float4*)(Aneigh + (long)gm * DIM) + c4);
    }
    stage_drain();
    __syncthreads();

    const int lane = tid & 31;
    const int wave = tid >> 5;                 // 0..15
    const int half = wave >> 3;                // 0 = self, 1 = neigh
    const int nt   = (wave & 7) * 16;          // N tile base within the half
    const int mr   = lane & 15;                // A: M = lane % 16 (ISA layout)
    const int nc   = lane & 15;                // B/C: N = lane % 16
    const int khi  = (lane >> 4) ? 2 : 0;      // lanes 16..31 hold K+2,K+3
    const float* __restrict__ brow  = (half ? WnT : WsT) + (long)(nt + nc) * DIM;
    const float* __restrict__ abase = ldsA + (half * MROWS + mr) * LDSW;

    v8f c[MT];
#pragma unroll
    for (int mt = 0; mt < MT; ++mt) c[mt] = (v8f){};

    for (int k = 0; k < DIM; k += 4) {
        const v2f b = *(const v2f*)(brow + k + khi);           // one b64 load
#pragma unroll
        for (int mt = 0; mt < MT; ++mt) {
            const v2f a = *(const v2f*)(abase + (mt * 16) * LDSW + k + khi);
            c[mt] = __builtin_amdgcn_wmma_f32_16x16x4_f32(false, a, false, b,
                                                          (short)0, c[mt],
                                                          false, false);
        }
    }

    // C layout: VGPR r -> M = r (lanes 0..15) / r+8 (lanes 16..31), N = lane%16
    const int mbase = (lane >> 4) ? 8 : 0;
#pragma unroll
    for (int mt = 0; mt < MT; ++mt) {
#pragma unroll
        for (int r = 0; r < 8; ++r) {
            const int gm = m0 + mt * 16 + r + mbase;
            float v = c[mt][r];
            if (doRelu) v = fmaxf(v, 0.0f);
            out[(long)gm * DIM + half * OUTC + nt + nc] = v;
        }
    }
}

// ---------------------------------------------------------------
// row-wise L2 normalize: out[m,:] = in[m,:] / max(||in[m,:]||, 1e-12)
// ---------------------------------------------------------------
__global__ __launch_bounds__(256)
void l2norm_kernel(const float* __restrict__ in, float* __restrict__ out, int M)
{
    __shared__ float red[8];
    const int m = blockIdx.x;
    const int t = threadIdx.x;
    if (m >= M) return;
    float v = in[(long)m * DIM + t];
    float s = v * v;
#pragma unroll
    for (int off = 16; off > 0; off >>= 1) s += __shfl_down(s, off, 32);
    if ((t & 31) == 0) red[t >> 5] = s;
    __syncthreads();
    float tot = red[0] + red[1] + red[2] + red[3] + red[4] + red[5] + red[6] + red[7];
    float nrm = fmaxf(sqrtf(tot), 1e-12f);
    out[(long)m * DIM + t] = v / nrm;
}

// ---------------------------------------------------------------
// aff[b] = dot(u1[b,:], u2[b,:])
// ---------------------------------------------------------------
__global__ __launch_bounds__(256)
void aff_kernel(const float* __restrict__ u1, const float* __restrict__ u2,
                float* __restrict__ aff)
{
    __shared__ float red[8];
    const int b = blockIdx.x;
    const int t = threadIdx.x;
    float s = u1[(long)b * DIM + t] * u2[(long)b * DIM + t];
#pragma unroll
    for (int off = 16; off > 0; off >>= 1) s += __shfl_down(s, off, 32);
    if ((t & 31) == 0) red[t >> 5] = s;
    __syncthreads();
    if (t == 0)
        aff[b] = red[0] + red[1] + red[2] + red[3] + red[4] + red[5] + red[6] + red[7];
}

// ---------------------------------------------------------------
// neg_aff = u1 @ un^T  :  [1024,256] x [256,64] via f32 WMMA.
// un rows are contiguous in k, so B fragments are natural b64 loads.
// ---------------------------------------------------------------
__global__ __launch_bounds__(128)
void negaff_kernel(const float* __restrict__ u1, const float* __restrict__ un,
                   float* __restrict__ out)
{
    const int m0   = blockIdx.x * 16;
    const int lane = threadIdx.x & 31;
    const int wave = threadIdx.x >> 5;         // 0..3
    const int nt   = wave * 16;
    const int mr   = lane & 15;
    const int nc   = lane & 15;
    const int khi  = (lane >> 4) ? 2 : 0;
    const float* __restrict__ arw = u1 + (long)(m0 + mr) * DIM;
    const float* __restrict__ brw = un + (long)(nt + nc) * DIM;   // B[k][n] = un[n][k]

    v8f c = {};
    for (int k = 0; k < DIM; k += 4) {
        const v2f a = *(const v2f*)(arw + k + khi);
        const v2f b = *(const v2f*)(brw + k + khi);
        c = __builtin_amdgcn_wmma_f32_16x16x4_f32(false, a, false, b,
                                                  (short)0, c, false, false);
    }
    const int mbase = (lane >> 4) ? 8 : 0;
#pragma unroll
    for (int r = 0; r < 8; ++r)
        out[(long)(m0 + r + mbase) * NEGS + nt + nc] = c[r];
}

// ---------------------------------------------------------------
// loss = ( sum softplus(-aff) + sum softplus(neg_aff) ) / B
// ---------------------------------------------------------------
__device__ __forceinline__ float softplus_f(float x)
{
    return fmaxf(x, 0.0f) + log1pf(expf(-fabsf(x)));
}

__global__ __launch_bounds__(256)
void loss_kernel(const float* __restrict__ aff, const float* __restrict__ negaff,
                 float* __restrict__ loss)
{
    __shared__ float red[8];
    const int t = threadIdx.x;
    float s = 0.f;
    for (int i = t; i < BATCH; i += 256)         s += softplus_f(-aff[i]);
    for (int i = t; i < BATCH * NEGS; i += 256)  s += softplus_f(negaff[i]);
#pragma unroll
    for (int off = 16; off > 0; off >>= 1) s += __shfl_down(s, off, 32);
    if ((t & 31) == 0) red[t >> 5] = s;
    __syncthreads();
    if (t == 0) {
        float tot = red[0] + red[1] + red[2] + red[3] + red[4] + red[5] + red[6] + red[7];
        loss[0] = tot / (float)BATCH;
    }
}

// ---------------------------------------------------------------
extern "C" void kernel_launch(void* const* d_in, const int* in_sizes, int n_in,
                              void* d_out, int out_size, void* d_ws, size_t ws_size,
                              hipStream_t stream)
{
    (void)in_sizes; (void)n_in; (void)out_size; (void)ws_size;

    const float* features = (const float*)d_in[0];
    const float* Ws0      = (const float*)d_in[1];
    const float* Wn0      = (const float*)d_in[2];
    const float* Ws1      = (const float*)d_in[3];
    const float* Wn1      = (const float*)d_in[4];
    const int*   s1_0 = (const int*)d_in[5];
    const int*   s1_1 = (const int*)d_in[6];
    const int*   s1_2 = (const int*)d_in[7];
    const int*   s2_0 = (const int*)d_in[8];
    const int*   s2_1 = (const int*)d_in[9];
    const int*   s2_2 = (const int*)d_in[10];
    const int*   n_0  = (const int*)d_in[11];
    const int*   n_1  = (const int*)d_in[12];
    const int*   n_2  = (const int*)d_in[13];

    float* outp = (float*)d_out;
    // d_out layout: [loss(1) | aff(B) | neg_aff(B*NEG) | u1(B*2*OUT)]
    float* o_loss = outp;
    float* o_aff  = outp + 1;
    float* o_neg  = outp + 1 + BATCH;
    float* o_u1   = outp + 1 + BATCH + BATCH * NEGS;

    // workspace layout (floats)
    float* ws  = (float*)d_ws;
    size_t off = 0;
    float* mh2  = ws + off; off += (size_t)BATCH * S2N * DIM;  // 10240 x 256
    float* g1   = ws + off; off += (size_t)BATCH * S2N * DIM;  // 10240 x 256
    float* mh1  = ws + off; off += (size_t)BATCH * DIM;
    float* g0   = ws + off; off += (size_t)BATCH * DIM;
    float* mg1  = ws + off; off += (size_t)BATCH * DIM;
    float* raw  = ws + off; off += (size_t)BATCH * DIM;
    float* u2   = ws + off; off += (size_t)BATCH * DIM;
    float* un   = ws + off; off += (size_t)NEGS * DIM;
    float* WsT0 = ws + off; off += (size_t)DIM * OUTC;
    float* WnT0 = ws + off; off += (size_t)DIM * OUTC;
    float* WsT1 = ws + off; off += (size_t)DIM * OUTC;
    float* WnT1 = ws + off; off += (size_t)DIM * OUTC;

    const size_t gemmLds = (size_t)2 * MROWS * LDSW * sizeof(float); // ~130 KB

    // one-time (per call) weight transposes, shared by all three encodes
    {
        dim3 tg(DIM / 16, OUTC / 16);
        transpose_w_kernel<<<tg, 256, 0, stream>>>(Ws0, WsT0);
        transpose_w_kernel<<<tg, 256, 0, stream>>>(Wn0, WnT0);
        transpose_w_kernel<<<tg, 256, 0, stream>>>(Ws1, WsT1);
        transpose_w_kernel<<<tg, 256, 0, stream>>>(Wn1, WnT1);
    }

    auto encode = [&](const int* i0, const int* i1, const int* i2, int bs, float* u_dst) {
        const int M1 = bs * S2N;
        // hop means (gather-bandwidth bound part)
        gather_mean_kernel<<<M1, 64, 0, stream>>>(features, i2, mh2, M1, S1N);
        gather_mean_kernel<<<bs, 64, 0, stream>>>(features, i1, mh1, bs, S2N);
        // layer 0 (ReLU)
        sage_gemm_kernel<<<M1 / MROWS, 512, gemmLds, stream>>>(features, i1, mh2, WsT0, WnT0, g1, M1, 1);
        sage_gemm_kernel<<<bs / MROWS, 512, gemmLds, stream>>>(features, i0, mh1, WsT0, WnT0, g0, bs, 1);
        // mean of g1 over S2 contiguous rows
        gather_mean_kernel<<<bs, 64, 0, stream>>>(g1, nullptr, mg1, bs, S2N);
        // layer 1 (identity)
        sage_gemm_kernel<<<bs / MROWS, 512, gemmLds, stream>>>(g0, nullptr, mg1, WsT1, WnT1, raw, bs, 0);
        l2norm_kernel<<<bs, 256, 0, stream>>>(raw, u_dst, bs);
    };

    encode(s1_0, s1_1, s1_2, BATCH, o_u1);   // u1 straight into d_out slot
    encode(s2_0, s2_1, s2_2, BATCH, u2);
    encode(n_0,  n_1,  n_2,  NEGS,  un);

    aff_kernel<<<BATCH, 256, 0, stream>>>(o_u1, u2, o_aff);
    negaff_kernel<<<BATCH / 16, 128, 0, stream>>>(o_u1, un, o_neg);
    loss_kernel<<<1, 256, 0, stream>>>(o_aff, o_neg, o_loss);
}